// HybridTwoTower_65506841198660
// MI455X (gfx1250) — compile-verified
//
#include <hip/hip_runtime.h>
#include <math.h>

#define N_ITEMS_C 200000
#define N_USERS_C 16384
#define N_HIST_C  819200
#define TOW 128
#define HID 256
#define EPS_F 1e-12f

typedef float v2f __attribute__((ext_vector_type(2)));
typedef float v8f __attribute__((ext_vector_type(8)));

// A fragment: 16x4 fp32 tile from LDS (row-major, ld = 256), M-tile base row mo.
// Per ISA 7.12.2: lanes 0-15 -> M=lane, {K=k,k+1}; lanes 16-31 -> M=lane-16, {K=k+2,k+3}.
__device__ __forceinline__ v2f frag_a(const float* x, int mo, int k, int lane) {
    int m = mo + (lane & 15);
    int koff = (lane < 16) ? 0 : 2;
    v2f a;
    a.x = x[m * 256 + k + koff];
    a.y = x[m * 256 + k + koff + 1];
    return a;
}

// B fragment: 4x16 fp32 tile from global weights (row-major K x N, leading dim ldn).
// lanes 0-15 -> N=lane, {K=k,k+1}; lanes 16-31 -> N=lane-16, {K=k+2,k+3}.
__device__ __forceinline__ v2f frag_b(const float* __restrict__ w, int k, int n0, int lane, int ldn) {
    int n = n0 + (lane & 15);
    int koff = (lane < 16) ? 0 : 2;
    v2f b;
    b.x = w[(k + koff) * ldn + n];
    b.y = w[(k + koff + 1) * ldn + n];
    return b;
}

// Fused tower: out[g] = l2norm(relu([inA[idx(g)] || inB[g]] @ w1 + b1) @ w2 + b2)
// 32 rows per block (2 M-tiles, B fragments reused across both), 256 threads = 8 waves.
__global__ __launch_bounds__(256) void tower_kernel(
    const float* __restrict__ inA, const int* __restrict__ idxA,
    const float* __restrict__ inB,
    const float* __restrict__ w1, const float* __restrict__ b1,
    const float* __restrict__ w2, const float* __restrict__ b2,
    float* __restrict__ out, int nRows)
{
    __shared__ float sx[32 * 256];   // input tile   (32 KB)
    __shared__ float sh[32 * 256];   // hidden tile  (32 KB)
    __shared__ float so[32 * 128];   // pre-norm out (16 KB)
    __shared__ float sred[32 * 8];
    __shared__ float sinv[32];

    const int tid  = threadIdx.x;
    const int lane = tid & 31;
    const int wave = tid >> 5;
    const int row0 = blockIdx.x * 32;

    // Stage X tile [32 x 256] = [inA || inB] into LDS (float4, zero-pad OOB rows).
    for (int idx = tid; idx < 32 * 64; idx += 256) {
        int r   = idx >> 6;
        int c4  = idx & 63;
        int col = c4 << 2;
        int g   = row0 + r;
        float4 v = make_float4(0.f, 0.f, 0.f, 0.f);
        if (g < nRows) {
            if (col < 128) {
                long src = idxA ? (long)idxA[g] : (long)g;
                v = *(const float4*)(inA + src * 128 + col);
            } else {
                v = *(const float4*)(inB + (long)g * 128 + (col - 128));
            }
        }
        *(float4*)(sx + r * 256 + col) = v;
    }
    __syncthreads();

    // Layer 1: hidden[32 x 256] = relu(X @ w1 + b1).
    // Wave w owns N-tiles {32w, 32w+16} x M-tiles {0, 16}; each B frag feeds 2 WMMAs.
    {
        int n0 = wave * 32;
        v8f a00 = {0.f,0.f,0.f,0.f,0.f,0.f,0.f,0.f};  // M-tile 0, N-tile 0
        v8f a01 = {0.f,0.f,0.f,0.f,0.f,0.f,0.f,0.f};  // M-tile 0, N-tile 1
        v8f a10 = {0.f,0.f,0.f,0.f,0.f,0.f,0.f,0.f};  // M-tile 1, N-tile 0
        v8f a11 = {0.f,0.f,0.f,0.f,0.f,0.f,0.f,0.f};  // M-tile 1, N-tile 1
        for (int k = 0; k < 256; k += 4) {
            v2f fa0 = frag_a(sx, 0,  k, lane);
            v2f fa1 = frag_a(sx, 16, k, lane);
            v2f fb0 = frag_b(w1, k, n0,      lane, 256);
            v2f fb1 = frag_b(w1, k, n0 + 16, lane, 256);
            a00 = __builtin_amdgcn_wmma_f32_16x16x4_f32(false, fa0, false, fb0, (short)0, a00, false, false);
            a10 = __builtin_amdgcn_wmma_f32_16x16x4_f32(false, fa1, false, fb0, (short)0, a10, false, false);
            a01 = __builtin_amdgcn_wmma_f32_16x16x4_f32(false, fa0, false, fb1, (short)0, a01, false, false);
            a11 = __builtin_amdgcn_wmma_f32_16x16x4_f32(false, fa1, false, fb1, (short)0, a11, false, false);
        }
        int mOff = (lane < 16) ? 0 : 8;
        int n    = lane & 15;
        float bb0 = b1[n0 + n];
        float bb1 = b1[n0 + 16 + n];
        for (int r = 0; r < 8; ++r) {
            int m = r + mOff;
            float v00 = a00[r] + bb0;
            float v01 = a01[r] + bb1;
            float v10 = a10[r] + bb0;
            float v11 = a11[r] + bb1;
            sh[m * 256 + n0 + n]             = v00 > 0.f ? v00 : 0.f;
            sh[m * 256 + n0 + 16 + n]        = v01 > 0.f ? v01 : 0.f;
            sh[(m + 16) * 256 + n0 + n]      = v10 > 0.f ? v10 : 0.f;
            sh[(m + 16) * 256 + n0 + 16 + n] = v11 > 0.f ? v11 : 0.f;
        }
    }
    __syncthreads();

    // Layer 2: out[32 x 128] = hidden @ w2 + b2. Wave w owns N-tile 16w x M-tiles {0,16}.
    {
        int n0 = wave * 16;
        v8f a0 = {0.f,0.f,0.f,0.f,0.f,0.f,0.f,0.f};
        v8f a1 = {0.f,0.f,0.f,0.f,0.f,0.f,0.f,0.f};
        for (int k = 0; k < 256; k += 4) {
            v2f fa0 = frag_a(sh, 0,  k, lane);
            v2f fa1 = frag_a(sh, 16, k, lane);
            v2f fb  = frag_b(w2, k, n0, lane, 128);
            a0 = __builtin_amdgcn_wmma_f32_16x16x4_f32(false, fa0, false, fb, (short)0, a0, false, false);
            a1 = __builtin_amdgcn_wmma_f32_16x16x4_f32(false, fa1, false, fb, (short)0, a1, false, false);
        }
        int mOff = (lane < 16) ? 0 : 8;
        int n    = lane & 15;
        float bb = b2[n0 + n];
        for (int r = 0; r < 8; ++r) {
            int m = r + mOff;
            so[m * 128 + n0 + n]        = a0[r] + bb;
            so[(m + 16) * 128 + n0 + n] = a1[r] + bb;
        }
    }
    __syncthreads();

    // Row-wise L2 norm: 8 threads per row, each sums 16 squares.
    {
        int r   = tid >> 3;   // 0..31
        int sub = tid & 7;    // 0..7
        float p = 0.f;
        for (int i = 0; i < 16; ++i) {
            float v = so[r * 128 + sub + 8 * i];
            p += v * v;
        }
        sred[r * 8 + sub] = p;
    }
    __syncthreads();
    if (tid < 32) {
        float s = 0.f;
        for (int i = 0; i < 8; ++i) s += sred[tid * 8 + i];
        sinv[tid] = 1.f / fmaxf(sqrtf(s), EPS_F);
    }
    __syncthreads();

    // Vectorized normalized store.
    for (int idx = tid; idx < 32 * 32; idx += 256) {   // 1024 float4s
        int r   = idx >> 5;
        int col = (idx & 31) << 2;
        int g   = row0 + r;
        if (g < nRows) {
            float inv = sinv[r];
            float4 v = *(const float4*)(so + r * 128 + col);
            v.x *= inv; v.y *= inv; v.z *= inv; v.w *= inv;
            *(float4*)(out + (long)g * 128 + col) = v;
        }
    }
}

__global__ void zero_i32(int* p, int n) {
    int i = blockIdx.x * blockDim.x + threadIdx.x;
    if (i < n) p[i] = 0;
}

// segids are sorted: mark [start,end) of each user's history slice.
__global__ void seg_bounds(const int* __restrict__ segids, int* __restrict__ start,
                           int* __restrict__ end, int nhist) {
    int i = blockIdx.x * blockDim.x + threadIdx.x;
    if (i >= nhist) return;
    int s = segids[i];
    if (i == 0 || segids[i - 1] != s) start[s] = i;
    if (i == nhist - 1 || segids[i + 1] != s) end[s] = i + 1;
}

// One 128-thread block per user: deterministic mean over its history rows.
__global__ __launch_bounds__(128) void pool_kernel(
    const float* __restrict__ item_emb, const int* __restrict__ hist_items,
    const int* __restrict__ start, const int* __restrict__ end,
    float* __restrict__ hist)
{
    int u = blockIdx.x;
    int c = threadIdx.x;
    int s = start[u], e = end[u];
    float sum = 0.f;
    for (int j = s; j < e; ++j) {
        long it = (long)hist_items[j];
        sum += item_emb[it * 128 + c];
    }
    int cnt = e - s;
    hist[(long)u * 128 + c] = (cnt > 0) ? (sum / (float)cnt) : 0.f;
}

extern "C" void kernel_launch(void* const* d_in, const int* in_sizes, int n_in,
                              void* d_out, int out_size, void* d_ws, size_t ws_size,
                              hipStream_t stream) {
    const float* item_text = (const float*)d_in[0];
    const float* gcn_item  = (const float*)d_in[1];
    const float* gcn_user  = (const float*)d_in[2];
    const float* w1_item   = (const float*)d_in[3];
    const float* b1_item   = (const float*)d_in[4];
    const float* w2_item   = (const float*)d_in[5];
    const float* b2_item   = (const float*)d_in[6];
    const float* w1_user   = (const float*)d_in[7];
    const float* b1_user   = (const float*)d_in[8];
    const float* w2_user   = (const float*)d_in[9];
    const float* b2_user   = (const float*)d_in[10];
    const int*   users     = (const int*)d_in[11];
    const int*   hist_it   = (const int*)d_in[12];
    const int*   hist_seg  = (const int*)d_in[13];

    float* user_emb = (float*)d_out;                       // [16384 x 128]
    float* item_emb = user_emb + (size_t)N_USERS_C * TOW;  // [200000 x 128]

    char*  ws       = (char*)d_ws;
    float* hist     = (float*)ws;                                              // 8 MB
    int*   segstart = (int*)(ws + (size_t)N_USERS_C * TOW * sizeof(float));
    int*   segend   = segstart + N_USERS_C;

    // 1) zero segment bounds (users with empty history -> count 0 -> zero row)
    zero_i32<<<(2 * N_USERS_C + 255) / 256, 256, 0, stream>>>(segstart, 2 * N_USERS_C);

    // 2) item tower -> item_emb (written directly into d_out tail)
    tower_kernel<<<(N_ITEMS_C + 31) / 32, 256, 0, stream>>>(
        item_text, nullptr, gcn_item,
        w1_item, b1_item, w2_item, b2_item,
        item_emb, N_ITEMS_C);

    // 3) segment boundaries from sorted segids
    seg_bounds<<<(N_HIST_C + 255) / 256, 256, 0, stream>>>(hist_seg, segstart, segend, N_HIST_C);

    // 4) per-user mean pooling of item embeddings
    pool_kernel<<<N_USERS_C, 128, 0, stream>>>(item_emb, hist_it, segstart, segend, hist);

    // 5) user tower (gather gcn_user rows via users[]) -> user_emb
    tower_kernel<<<(N_USERS_C + 31) / 32, 256, 0, stream>>>(
        gcn_user, users, hist,
        w1_user, b1_user, w2_user, b2_user,
        user_emb, N_USERS_C);
}